// UnbalancedMMSE_33749853012537
// MI455X (gfx1250) — compile-verified
//
#include <hip/hip_runtime.h>

typedef float v8f __attribute__((ext_vector_type(8)));
typedef float v2f __attribute__((ext_vector_type(2)));

#define N_PTS 2048
#define DIMS 128
#define T_MAJ 512
#define T_TOT 1024
#define LDS_STRIDE 132   // 16 rows x 128 cols, padded (r*132 % 64 varies -> no bank conflicts)

// s_wait_asynccnt: builtin if available, else raw CDNA5 SOPP encoding via asm.
#if defined(__has_builtin) && __has_builtin(__builtin_amdgcn_s_wait_asynccnt)
#define WAIT_ASYNC(n) __builtin_amdgcn_s_wait_asynccnt(n)
#else
#define WAIT_ASYNC(n) asm volatile("s_wait_asynccnt %0" ::"i"(n) : "memory")
#endif

// One 32B/thread async global->LDS stage of a 16x128 f32 A tile (8 KB total for
// 256 threads). INST_OFFSET is added to BOTH the global and LDS address per the
// CDNA5 async pseudocode, so one base pair + offset:16 moves both 16B chunks.
__device__ __forceinline__ void stage_async(uint32_t lds_off, uint64_t gaddr) {
    asm volatile("global_load_async_to_lds_b128 %0, %1, off"
                 ::"v"(lds_off), "v"(gaddr) : "memory");
    asm volatile("global_load_async_to_lds_b128 %0, %1, off offset:16"
                 ::"v"(lds_off), "v"(gaddr) : "memory");
}

// ---------------------------------------------------------------------------
// K1: per (batch, 128-wide t-group) workgroup.
//  - 8 waves, each owns a 16-wide t-tile; B panel (128x16 f32) in 64 VGPRs.
//  - double-buffered LDS A tile staged with GLOBAL_LOAD_ASYNC_TO_LDS_B128
//    (ASYNCcnt), overlapped with the WMMA chain of the current tile.
//  - 32x v_wmma_f32_16x16x4_f32 split into two independent accumulation chains,
//    then in-register + shfl exclusive prefix-sum of (target - xw)^2 along n
//    with a running per-column carry (the cumsum dependency).
// ---------------------------------------------------------------------------
__global__ __launch_bounds__(256) void k1_gemm_cumsum(
    const float* __restrict__ data, const float* __restrict__ targets,
    const float* __restrict__ W, const float* __restrict__ Wm,
    float* __restrict__ pref, float* __restrict__ xw, int b0)
{
    __shared__ float Atile[2][16 * LDS_STRIDE];

    const int tid  = threadIdx.x;
    const int lane = tid & 31;
    const int wv   = tid >> 5;       // wave id 0..7
    const int g    = blockIdx.x;     // t-group 0..7 (128 tasks each)
    const int bb   = blockIdx.y;     // chunk-local batch
    const int b    = b0 + bb;        // global batch

    const int tbase = g * 128 + wv * 16;                 // global task base of this wave's tile
    const float* Wsel = (tbase < T_MAJ) ? W : Wm;        // major vs minor pool
    const int tcol  = (tbase & (T_MAJ - 1)) + (lane & 15);
    const int khalf = (lane >> 4) * 2;                   // A/B lane-half K offset

    // Preload B panel: 32 k-steps of 16x16x4 (K rows 4s+khalf, 4s+khalf+1)
    v2f bpan[32];
#pragma unroll
    for (int s = 0; s < 32; ++s) {
        const int k0 = 4 * s + khalf;
        bpan[s].x = Wsel[(size_t)k0 * T_MAJ + tcol];
        bpan[s].y = Wsel[(size_t)(k0 + 1) * T_MAJ + tcol];
    }

    const size_t browbase = (size_t)b * N_PTS;

    // Per-thread staging addresses: row = tid>>4, cols (tid&15)*8 .. +7
    const int srow = tid >> 4;
    const int scol = (tid & 15) * 8;
    const uint64_t gstage0 =
        (uint64_t)(uintptr_t)(data + (browbase + srow) * DIMS + scol);
    // Low 32 bits of a generic pointer to LDS == LDS byte offset (aperture is
    // in the high bits), which is what the async-load VDST operand wants.
    const uint32_t lstage[2] = {
        (uint32_t)(uintptr_t)&Atile[0][srow * LDS_STRIDE + scol],
        (uint32_t)(uintptr_t)&Atile[1][srow * LDS_STRIDE + scol]
    };

    // Prologue: stage tile 0 into buffer 0.
    stage_async(lstage[0], gstage0);

    float carry = 0.0f;                                  // running SSR for this lane's column

    for (int nt = 0; nt < N_PTS / 16; ++nt) {
        const int nbase = nt * 16;
        const int cur   = nt & 1;

        if (nt + 1 < N_PTS / 16) {
            // Stage next tile into the other buffer while we compute this one.
            stage_async(lstage[cur ^ 1],
                        gstage0 + (size_t)(nbase + 16) * DIMS * sizeof(float));
            WAIT_ASYNC(2);   // in-order: older tile's 2 transfers have landed
        } else {
            WAIT_ASYNC(0);
        }
        __syncthreads();     // whole A tile visible in LDS buffer cur

        // XW tile = A(16x128) x B(128x16): two independent f32 WMMA chains.
        v8f acc0 = {}, acc1 = {};
        const float* arow = &Atile[cur][(lane & 15) * LDS_STRIDE];
#pragma unroll
        for (int s = 0; s < 16; ++s) {
            v2f a0, a1;
            a0.x = arow[8 * s + khalf];
            a0.y = arow[8 * s + khalf + 1];
            a1.x = arow[8 * s + 4 + khalf];
            a1.y = arow[8 * s + 4 + khalf + 1];
            acc0 = __builtin_amdgcn_wmma_f32_16x16x4_f32(
                false, a0, false, bpan[2 * s], (short)0, acc0, false, false);
            acc1 = __builtin_amdgcn_wmma_f32_16x16x4_f32(
                false, a1, false, bpan[2 * s + 1], (short)0, acc1, false, false);
        }
        const v8f acc = acc0 + acc1;

        // Exclusive prefix sum of squared residuals along n within the tile.
        const int rh = (lane >> 4) * 8;                  // row offset of this lane half
        float d2[8], incl[8];
        float run = 0.0f;
#pragma unroll
        for (int r = 0; r < 8; ++r) {
            const float tgt = targets[browbase + nbase + rh + r];
            const float d   = tgt - acc[r];
            d2[r]   = d * d;
            run    += d2[r];
            incl[r] = run;                               // inclusive within this half
        }
        const float lowtot  = __shfl(incl[7], lane & 15, 32);        // rows 0..7 total
        const float addhi   = (lane >= 16) ? lowtot : 0.0f;
        const float tiletot = __shfl(incl[7] + addhi, (lane & 15) | 16, 32); // rows 0..15

        const size_t outrow0 =
            ((size_t)bb * N_PTS + nbase + rh) * T_TOT + tbase + (lane & 15);
#pragma unroll
        for (int r = 0; r < 8; ++r) {
            const size_t idx = outrow0 + (size_t)r * T_TOT;
            xw[idx]   = acc[r];
            pref[idx] = carry + addhi + incl[r] - d2[r]; // exclusive prefix
        }
        carry += tiletot;
        __syncthreads();     // all reads of LDS buffer cur done before it is restaged
    }
}

// ---------------------------------------------------------------------------
// K2: one wave32 per (b, n) row: softmax over the 1024 tasks of
//     (-0.5 * prefix + log_prior), then pred = sum_t w_t * xw_t.
//     (Second einsum eliminated algebraically: preds = sum_t w_t * XW_t.)
// ---------------------------------------------------------------------------
__global__ __launch_bounds__(256) void k2_softmax_pred(
    const float* __restrict__ pref, const float* __restrict__ xw,
    float* __restrict__ out, int b0)
{
    const int lane = threadIdx.x & 31;
    const int wv   = threadIdx.x >> 5;
    const size_t row  = (size_t)blockIdx.x * 8 + wv;     // chunk-local (b,n) row
    const size_t base = row * T_TOT;

    const float lpM = -1.2039728043259361f;  // log((1-0.1)/3)
    const float lpm = -8.5409097180335540f;  // log(0.1/512)

    float lp[32], xv[32];
    float m = -3.4e38f;
#pragma unroll
    for (int i = 0; i < 32; ++i) {
        const int t = i * 32 + lane;
        const float p = pref[base + t];
        xv[i] = xw[base + t];
        const float v = -0.5f * p + ((t < T_MAJ) ? lpM : lpm);
        lp[i] = v;
        m = fmaxf(m, v);
    }
#pragma unroll
    for (int msk = 16; msk; msk >>= 1)
        m = fmaxf(m, __shfl_xor(m, msk, 32));

    float sw = 0.0f, swx = 0.0f;
#pragma unroll
    for (int i = 0; i < 32; ++i) {
        const float e = __expf(lp[i] - m);
        sw  += e;
        swx += e * xv[i];
    }
#pragma unroll
    for (int msk = 16; msk; msk >>= 1) {
        sw  += __shfl_xor(sw,  msk, 32);
        swx += __shfl_xor(swx, msk, 32);
    }
    if (lane == 0)
        out[(size_t)b0 * N_PTS + row] = swx / sw;
}

// ---------------------------------------------------------------------------
extern "C" void kernel_launch(void* const* d_in, const int* in_sizes, int n_in,
                              void* d_out, int out_size, void* d_ws, size_t ws_size,
                              hipStream_t stream) {
    (void)in_sizes; (void)n_in; (void)out_size;

    const float* data    = (const float*)d_in[0];   // (64, 2048, 128)
    const float* targets = (const float*)d_in[1];   // (64, 2048)
    const float* W       = (const float*)d_in[2];   // (128, 512)
    const float* Wm      = (const float*)d_in[3];   // (128, 512)
    float* out = (float*)d_out;                     // (64, 2048)

    const int B = 64;
    // Scratch per batch: prefix + xw, f32 each: N*T_TOT*8 bytes = 16 MB.
    const size_t per_b = (size_t)N_PTS * T_TOT * 2 * sizeof(float);
    int bc = (int)(ws_size / per_b);
    if (bc < 1) bc = 1;
    if (bc > B) bc = B;

    float* prefbuf = (float*)d_ws;
    float* xwbuf   = prefbuf + (size_t)bc * N_PTS * T_TOT;

    for (int b0 = 0; b0 < B; b0 += bc) {
        const int bcur = (B - b0 < bc) ? (B - b0) : bc;
        dim3 g1(8, bcur);  // 8 t-groups x batches
        k1_gemm_cumsum<<<g1, 256, 0, stream>>>(data, targets, W, Wm, prefbuf, xwbuf, b0);
        k2_softmax_pred<<<bcur * (N_PTS / 8), 256, 0, stream>>>(prefbuf, xwbuf, out, b0);
    }
}